// LogMap_78778290143327
// MI455X (gfx1250) — compile-verified
//
#include <hip/hip_runtime.h>
#include <math.h>

// Batched SPD matrix log via inverse scaling-and-squaring + Newton-Schulz,
// all heavy work as 64x64x64 fp32 GEMMs on V_WMMA_F32_16X16X4_F32.
// One workgroup (8 waves) per matrix; 4 LDS ping-pong buffers.
//
// All intermediate matrices are symmetric (polynomials in SPD X), so B-matrix
// WMMA fragments are read via B[k][n] == B[n][k] as contiguous row pairs ->
// ds_load_b64, same as A. LDS rows padded to 68 dwords so every fragment
// load/store is LDS bank-conflict-free (bank = 4*row + col over 64 banks).

typedef float v2f __attribute__((ext_vector_type(2)));
typedef float v8f __attribute__((ext_vector_type(8)));

#define CDIM 64          // matrix dimension (global layout stride)
#define SDIM 68          // padded LDS leading dimension (dwords)
#define NTHREADS 256

// Epilogue modes fused into the GEMM store.
#define EPI_COPY   0  // D = acc
#define EPI_NST    1  // D = 1.5 I - 0.5 acc        (Newton-Schulz T matrix)
#define EPI_DIAG   2  // D = acc + cdiag * I        (Horner step)
#define EPI_FINAL  3  // D = 16 acc + cdiag * I     (final global store)

// D = A * B (64x64 fp32, A/B symmetric, LDS stride SDIM; D stride dld).
// Wave w owns row-block ti=w>>1 and col-blocks tj0=(w&1)*2, tj0+1 -> A
// fragments shared by both output tiles. Fragment layouts per ISA 7.12.2:
//   A 16x4 f32 : lane L -> M = L&15 ; VGPRs = K+kh, K+kh+1, kh=(L>=16)*2
//   B 4x16 f32 : lane L -> N = L&15 ; VGPRs = K+kh, K+kh+1  (read as B^T row)
//   C/D 16x16  : VGPR r -> M = r + (L>=16)*8 ; N = L&15
template <int MODE>
__device__ __forceinline__ void mm64(const float* __restrict__ A,
                                     const float* __restrict__ B,
                                     float* __restrict__ D, int dld,
                                     int lane, int wave, float cdiag) {
  const int ti  = wave >> 1;
  const int tj0 = (wave & 1) * 2;
  const int m   = lane & 15;
  const int kh  = (lane >> 4) << 1;         // 0 or 2
  const int row = ti * 16 + m;
  const int n0  = tj0 * 16 + m;

  // All fragments are contiguous (K,K+1) row pairs -> ds_load_(2addr_)b64.
  v2f afrag[16], bfrag0[16], bfrag1[16];
#pragma unroll
  for (int k = 0; k < 16; ++k) {
    const int kb = k * 4 + kh;
    afrag[k]  = *(const v2f*)&A[row * SDIM + kb];
    bfrag0[k] = *(const v2f*)&B[n0 * SDIM + kb];         // B symmetric
    bfrag1[k] = *(const v2f*)&B[(n0 + 16) * SDIM + kb];  // B symmetric
  }

  v8f acc0 = {}, acc1 = {};
#pragma unroll
  for (int k = 0; k < 16; ++k) {
    acc0 = __builtin_amdgcn_wmma_f32_16x16x4_f32(false, afrag[k], false,
                                                 bfrag0[k], (short)0, acc0,
                                                 false, false);
    acc1 = __builtin_amdgcn_wmma_f32_16x16x4_f32(false, afrag[k], false,
                                                 bfrag1[k], (short)0, acc1,
                                                 false, false);
  }

  const int mbase = ti * 16 + ((lane >> 4) << 3);
#pragma unroll
  for (int r = 0; r < 8; ++r) {
    const int rr = mbase + r;
    float v0 = acc0[r], v1 = acc1[r];
    if (MODE == EPI_NST) {
      v0 = -0.5f * v0; v1 = -0.5f * v1;
      if (rr == n0)      v0 += 1.5f;
      if (rr == n0 + 16) v1 += 1.5f;
    } else if (MODE == EPI_DIAG) {
      if (rr == n0)      v0 += cdiag;
      if (rr == n0 + 16) v1 += cdiag;
    } else if (MODE == EPI_FINAL) {
      v0 *= 16.f; v1 *= 16.f;
      if (rr == n0)      v0 += cdiag;
      if (rr == n0 + 16) v1 += cdiag;
    }
    D[rr * dld + n0]      = v0;
    D[rr * dld + n0 + 16] = v1;
  }
}

__global__ __launch_bounds__(NTHREADS)
void logmap_spd64_kernel(const float* __restrict__ X, float* __restrict__ out,
                         int nmat) {
  __shared__ float sA[CDIM * SDIM];  // Y
  __shared__ float sB[CDIM * SDIM];  // Z
  __shared__ float sT[CDIM * SDIM];  // T / Horner R
  __shared__ float sW[CDIM * SDIM];  // scratch / reduction

  const int tid  = threadIdx.x;
  const int lane = tid & 31;
  const int wave = tid >> 5;
  const int mat  = blockIdx.x;
  if (mat >= nmat) return;  // block-uniform

  const float* Xg = X + (size_t)mat * CDIM * CDIM;
  float* Og       = out + (size_t)mat * CDIM * CDIM;

  // Load X into LDS (coalesced reads; padded rows in LDS).
  for (int i = tid; i < CDIM * CDIM; i += NTHREADS)
    sA[(i >> 6) * SDIM + (i & 63)] = Xg[i];
  __syncthreads();

  // alpha = ||X||_inf >= lambda_max  (rowsums in sW, then max into sT[0..1]).
  if (tid < CDIM) {
    float s = 0.f;
    for (int c = 0; c < CDIM; ++c) s += fabsf(sA[tid * SDIM + c]);
    sW[tid] = s;
  }
  __syncthreads();
  if (tid == 0) {
    float a = sW[0];
    for (int r = 1; r < CDIM; ++r) a = fmaxf(a, sW[r]);
    sT[0] = 1.0f / a;
    sT[1] = logf(a);
  }
  __syncthreads();
  const float inva = sT[0];
  const float loga = sT[1];  // latch before sT is recycled
  __syncthreads();

  // Y = X/alpha (spectrum in (0,1]), Z = I.
  for (int i = tid; i < CDIM * CDIM; i += NTHREADS) {
    const int r = i >> 6, c = i & 63;
    sA[r * SDIM + c] *= inva;
    sB[r * SDIM + c] = (r == c) ? 1.f : 0.f;
  }
  __syncthreads();

  float* Y = sA; float* Z = sB; float* T = sT; float* W = sW;

  // m = 4 coupled Newton-Schulz sqrt levels: Y -> (X/alpha)^(1/16).
  const int iters[4] = {10, 8, 7, 6};
#pragma unroll 1
  for (int lvl = 0; lvl < 4; ++lvl) {
    if (lvl > 0) {
      for (int i = tid; i < CDIM * CDIM; i += NTHREADS) {
        const int r = i >> 6, c = i & 63;
        Z[r * SDIM + c] = (r == c) ? 1.f : 0.f;
      }
      __syncthreads();
    }
#pragma unroll 1
    for (int it = 0; it < iters[lvl]; ++it) {
      mm64<EPI_NST>(Z, Y, T, SDIM, lane, wave, 0.f);   // T = 1.5I - 0.5 Z*Y
      __syncthreads();
      mm64<EPI_COPY>(Y, T, W, SDIM, lane, wave, 0.f);  // Y' = Y*T
      __syncthreads();
      { float* tmp = Y; Y = W; W = tmp; }
      mm64<EPI_COPY>(T, Z, W, SDIM, lane, wave, 0.f);  // Z' = T*Z
      __syncthreads();
      { float* tmp = Z; Z = W; W = tmp; }
    }
  }

  // E = Y - I, spectrum in [-0.13, 0] -> fast Taylor for log(I+E).
  for (int i = tid; i < CDIM * CDIM; i += NTHREADS) {
    const int r = i >> 6, c = i & 63;
    Z[r * SDIM + c] = Y[r * SDIM + c] - ((r == c) ? 1.f : 0.f);
  }
  __syncthreads();
  float* E = Z;

  // Horner for log(I+E) = E*(c1 I + c2 E + ... + cK E^(K-1)), c_k = (-1)^(k+1)/k.
  const int K = 12;
  {
    const float cK   = (K % 2 == 0) ? (-1.f / K) : (1.f / K);
    const float cKm1 = ((K - 1) % 2 == 0) ? (-1.f / (K - 1)) : (1.f / (K - 1));
    for (int i = tid; i < CDIM * CDIM; i += NTHREADS) {
      const int r = i >> 6, c = i & 63;
      float v = cK * E[r * SDIM + c];
      if (r == c) v += cKm1;
      T[r * SDIM + c] = v;
    }
  }
  __syncthreads();
  float* R = T;
#pragma unroll 1
  for (int j = K - 2; j >= 1; --j) {
    const float cj = (j & 1) ? (1.f / j) : (-1.f / j);
    mm64<EPI_DIAG>(E, R, W, SDIM, lane, wave, cj);     // R' = E*R + cj I
    __syncthreads();
    { float* tmp = R; R = W; W = tmp; }
  }

  // log X = log(alpha) I + 16 * (E*R), fused straight to global memory.
  mm64<EPI_FINAL>(E, R, Og, CDIM, lane, wave, loga);
}

extern "C" void kernel_launch(void* const* d_in, const int* in_sizes, int n_in,
                              void* d_out, int out_size, void* d_ws, size_t ws_size,
                              hipStream_t stream) {
  (void)n_in; (void)d_ws; (void)ws_size; (void)out_size;
  const float* X = (const float*)d_in[0];
  float* out = (float*)d_out;
  const int nmat = in_sizes[0] / (CDIM * CDIM);  // N*F = 6400 matrices
  logmap_spd64_kernel<<<nmat, NTHREADS, 0, stream>>>(X, out, nmat);
}